// Explainer_61057255080611
// MI455X (gfx1250) — compile-verified
//
#include <hip/hip_runtime.h>
#include <math.h>

#define NN 100000
#define NE 1250000
#define CH 64
#define ED 16

typedef __attribute__((ext_vector_type(2))) float v2f;
typedef __attribute__((ext_vector_type(8))) float v8f;

__device__ __forceinline__ v8f wmma_f32(v2f a, v2f b, v8f c) {
  // D = A(16x4,f32) * B(4x16,f32) + C(16x16,f32)
  return __builtin_amdgcn_wmma_f32_16x16x4_f32(false, a, false, b, (short)0, c,
                                               false, false);
}

__device__ __forceinline__ v8f splat8(float v) {
  v8f r;
#pragma unroll
  for (int i = 0; i < 8; ++i) r[i] = v;
  return r;
}

// ---------------------------------------------------------------- zero agg
__global__ __launch_bounds__(256) void zero_kernel(float4* __restrict__ p, int n4) {
  int i = blockIdx.x * 256 + threadIdx.x;
  if (i < n4) p[i] = make_float4(0.f, 0.f, 0.f, 0.f);
}

// ------------------------------------------------- edge transform + scatter
// e = edge_attr @ We + be ; m = relu(x[src] + e) ; atomic agg[dst] += m
// One wave = 16 edges/iter. B fragments in registers; gathers issued before
// the WMMA chain so their L2 latency is hidden under the matrix math.
__global__ __launch_bounds__(256) void edge_kernel(
    const float* __restrict__ x, const int* __restrict__ ei,
    const float* __restrict__ ea, const float* __restrict__ We,
    const float* __restrict__ be, float* __restrict__ agg) {
  __shared__ float sWe[ED * CH];  // 4 KB
  const int tid = threadIdx.x;
  for (int i = tid; i < ED * CH; i += 256) sWe[i] = We[i];
  __syncthreads();

  const int wave = tid >> 5;
  const int lane = tid & 31;
  const int row = lane & 15;   // M index (edge-local) / channel-low
  const int hi  = lane >> 4;
  const int kb  = hi * 2;      // K pair select per half-wave

  // hoist all B fragments (16 v2f = 32 VGPRs) + bias out of the tile loop
  v2f bf[4][4];
  float bev[4];
#pragma unroll
  for (int j = 0; j < 4; ++j) {
    const int c = 16 * j + row;
    bev[j] = be[c];
#pragma unroll
    for (int k4 = 0; k4 < 4; ++k4) {
      bf[k4][j].x = sWe[(4 * k4 + kb) * CH + c];
      bf[k4][j].y = sWe[(4 * k4 + kb + 1) * CH + c];
    }
  }

  const int stride = gridDim.x * 8;
  for (int tile = blockIdx.x * 8 + wave; tile < NE / 16; tile += stride) {
    const int e0 = tile * 16;

    // ---- phase 1: indices + all 32 gather loads issued up front ----------
    const int sv = ei[e0 + row];
    const int dv = ei[NE + e0 + row];
    int srcn[8], dstn[8];
#pragma unroll
    for (int r = 0; r < 8; ++r) {
      srcn[r] = __shfl(sv, r + hi * 8, 32);
      dstn[r] = __shfl(dv, r + hi * 8, 32);
    }
    float g[8][4];
#pragma unroll
    for (int r = 0; r < 8; ++r) {
      const float* xs = x + (size_t)srcn[r] * CH + row;
#pragma unroll
      for (int j = 0; j < 4; ++j) g[r][j] = xs[16 * j];
    }

    // ---- phase 2: GEMM (overlaps with in-flight gathers) -----------------
    v8f acc[4];
#pragma unroll
    for (int j = 0; j < 4; ++j) acc[j] = splat8(bev[j]);
    const float* arow = ea + (size_t)(e0 + row) * ED;
    v2f afrag[4];
#pragma unroll
    for (int k4 = 0; k4 < 4; ++k4)
      afrag[k4] = *(const v2f*)(arow + 4 * k4 + kb);
#pragma unroll
    for (int k4 = 0; k4 < 4; ++k4)
#pragma unroll
      for (int j = 0; j < 4; ++j) acc[j] = wmma_f32(afrag[k4], bf[k4][j], acc[j]);

    // ---- phase 3: relu + scatter atomics (no-return, no waits between) ---
#pragma unroll
    for (int r = 0; r < 8; ++r) {
      float* ad = agg + (size_t)dstn[r] * CH + row;
#pragma unroll
      for (int j = 0; j < 4; ++j) {
        float m = acc[j][r] + g[r][j];
        m = m > 0.f ? m : 0.f;
        __hip_atomic_fetch_add(ad + 16 * j, m, __ATOMIC_RELAXED,
                               __HIP_MEMORY_SCOPE_AGENT);
      }
    }
  }
}

// ---------------------------- fused node MLP: h=(1+eps)x+agg -> 2x GEMM -> LN
#define HPAD 66  // padded LDS stride: conflict-free + 8B-aligned float2
__global__ __launch_bounds__(128) void mlp_kernel(
    const float* __restrict__ xin, const float* __restrict__ agg,
    const float* __restrict__ W1, const float* __restrict__ b1,
    const float* __restrict__ W2, const float* __restrict__ b2,
    const float* __restrict__ epsl, const float* __restrict__ gamma,
    const float* __restrict__ beta, float* __restrict__ xout) {
  __shared__ float sW1[CH * CH];      // 16 KB
  __shared__ float sW2[CH * CH];      // 16 KB
  __shared__ float sH[4][16 * HPAD];  // per-wave h1 tile (wave-private)
  const int tid = threadIdx.x;
  for (int i = tid; i < CH * CH; i += 128) { sW1[i] = W1[i]; sW2[i] = W2[i]; }
  __syncthreads();

  const int wave = tid >> 5;
  const int lane = tid & 31;
  const int row = lane & 15, hi = lane >> 4, kb = hi * 2;
  float* H = sH[wave];
  const float e1 = 1.0f + epsl[0];

  float b1v[4], b2v[4], gv[4], btv[4];
#pragma unroll
  for (int j = 0; j < 4; ++j) {
    const int c = 16 * j + row;
    b1v[j] = b1[c];
    b2v[j] = b2[c];
    gv[j] = gamma[c];
    btv[j] = beta[c];
  }

  const int stride = gridDim.x * 4;
  for (int tile = blockIdx.x * 4 + wave; tile < NN / 16; tile += stride) {
    const int R = tile * 16;
    // ---------------- GEMM1: relu(((1+eps)x + agg) @ W1 + b1) ----------------
    v8f acc[4];
#pragma unroll
    for (int j = 0; j < 4; ++j) acc[j] = splat8(b1v[j]);
    const float* xrow = xin + (size_t)(R + row) * CH;
    const float* grow = agg + (size_t)(R + row) * CH;
#pragma unroll
    for (int kk = 0; kk < CH; kk += 4) {
      v2f xa = *(const v2f*)(xrow + kk + kb);
      v2f ga = *(const v2f*)(grow + kk + kb);
      v2f a;
      a.x = e1 * xa.x + ga.x;
      a.y = e1 * xa.y + ga.y;
#pragma unroll
      for (int j = 0; j < 4; ++j) {
        const int c = 16 * j + row;
        v2f b;
        b.x = sW1[(kk + kb) * CH + c];
        b.y = sW1[(kk + kb + 1) * CH + c];
        acc[j] = wmma_f32(a, b, acc[j]);
      }
    }
    // relu(h1) -> wave-private LDS tile for re-layout into A fragments
#pragma unroll
    for (int r = 0; r < 8; ++r)
#pragma unroll
      for (int j = 0; j < 4; ++j) {
        float v = acc[j][r];
        H[(r + hi * 8) * HPAD + 16 * j + row] = v > 0.f ? v : 0.f;
      }
    __builtin_amdgcn_wave_barrier();  // keep DS store->load order (same wave)

    // ---------------- GEMM2: h1 @ W2 + b2, then LayerNorm + relu -----------
    v8f acc2[4];
#pragma unroll
    for (int j = 0; j < 4; ++j) acc2[j] = splat8(b2v[j]);
#pragma unroll
    for (int kk = 0; kk < CH; kk += 4) {
      v2f a = *(const v2f*)(H + row * HPAD + kk + kb);
#pragma unroll
      for (int j = 0; j < 4; ++j) {
        const int c = 16 * j + row;
        v2f b;
        b.x = sW2[(kk + kb) * CH + c];
        b.y = sW2[(kk + kb + 1) * CH + c];
        acc2[j] = wmma_f32(a, b, acc2[j]);
      }
    }
    __builtin_amdgcn_wave_barrier();
#pragma unroll
    for (int r = 0; r < 8; ++r) {
      float s = 0.f, sq = 0.f;
#pragma unroll
      for (int j = 0; j < 4; ++j) {
        float v = acc2[j][r];
        s += v;
        sq += v * v;
      }
#pragma unroll
      for (int m = 1; m < 16; m <<= 1) {
        s += __shfl_xor(s, m, 32);
        sq += __shfl_xor(sq, m, 32);
      }
      const float mean = s * (1.f / 64.f);
      const float var = sq * (1.f / 64.f) - mean * mean;
      const float rstd = rsqrtf(var + 1e-5f);
      float* orow = xout + (size_t)(R + r + hi * 8) * CH;
#pragma unroll
      for (int j = 0; j < 4; ++j) {
        float v = (acc2[j][r] - mean) * rstd * gv[j] + btv[j];
        orow[16 * j + row] = v > 0.f ? v : 0.f;
      }
    }
  }
}

// --------------------------------- head: sigmoid(relu(x@Wh1+bh1)@Wh2 + bh2)
__global__ __launch_bounds__(128) void head_kernel(
    const float* __restrict__ xin, const float* __restrict__ Wh1,
    const float* __restrict__ bh1, const float* __restrict__ Wh2,
    const float* __restrict__ bh2, float* __restrict__ out) {
  __shared__ float sW[CH * CH];
  const int tid = threadIdx.x;
  for (int i = tid; i < CH * CH; i += 128) sW[i] = Wh1[i];
  __syncthreads();

  const int wave = tid >> 5;
  const int lane = tid & 31;
  const int row = lane & 15, hi = lane >> 4, kb = hi * 2;

  float bh1v[4], w2[4];
#pragma unroll
  for (int j = 0; j < 4; ++j) {
    bh1v[j] = bh1[16 * j + row];
    w2[j] = Wh2[16 * j + row];
  }
  const float b2v = bh2[0];

  const int stride = gridDim.x * 4;
  for (int tile = blockIdx.x * 4 + wave; tile < NN / 16; tile += stride) {
    const int R = tile * 16;
    v8f acc[4];
#pragma unroll
    for (int j = 0; j < 4; ++j) acc[j] = splat8(bh1v[j]);
    const float* xrow = xin + (size_t)(R + row) * CH;
#pragma unroll
    for (int kk = 0; kk < CH; kk += 4) {
      v2f a = *(const v2f*)(xrow + kk + kb);
#pragma unroll
      for (int j = 0; j < 4; ++j) {
        const int c = 16 * j + row;
        v2f b;
        b.x = sW[(kk + kb) * CH + c];
        b.y = sW[(kk + kb + 1) * CH + c];
        acc[j] = wmma_f32(a, b, acc[j]);
      }
    }
#pragma unroll
    for (int r = 0; r < 8; ++r) {
      float s = 0.f;
#pragma unroll
      for (int j = 0; j < 4; ++j) {
        float v = acc[j][r];
        v = v > 0.f ? v : 0.f;
        s += v * w2[j];
      }
#pragma unroll
      for (int m = 1; m < 16; m <<= 1) s += __shfl_xor(s, m, 32);
      if (row == 0) {
        const float z = s + b2v;
        out[R + r + hi * 8] = 1.f / (1.f + __expf(-z));
      }
    }
  }
}

// ------------------------------------------------------------------- launch
extern "C" void kernel_launch(void* const* d_in, const int* in_sizes, int n_in,
                              void* d_out, int out_size, void* d_ws,
                              size_t ws_size, hipStream_t stream) {
  const float* x     = (const float*)d_in[0];
  const int*   ei    = (const int*)d_in[1];
  const float* ea    = (const float*)d_in[2];
  const float* We    = (const float*)d_in[3];
  const float* be    = (const float*)d_in[4];
  const float* W1    = (const float*)d_in[5];
  const float* b1    = (const float*)d_in[6];
  const float* W2    = (const float*)d_in[7];
  const float* b2    = (const float*)d_in[8];
  const float* eps   = (const float*)d_in[9];
  const float* gamma = (const float*)d_in[10];
  const float* beta  = (const float*)d_in[11];
  const float* Wh1   = (const float*)d_in[12];
  const float* bh1   = (const float*)d_in[13];
  const float* Wh2   = (const float*)d_in[14];
  const float* bh2   = (const float*)d_in[15];
  float* out = (float*)d_out;

  float* bufA = (float*)d_ws;                      // [N, C]
  float* bufB = bufA + (size_t)NN * CH;            // [N, C]
  float* agg  = bufB + (size_t)NN * CH;            // [N, C]

  const int n4      = (NN * CH) / 4;
  const int zblocks = (n4 + 255) / 256;
  const int eblocks = 1024;  // grid-stride: 8192 wave streams over 78125 tiles
  const int nblocks = 512;   // grid-stride: 2048 wave streams over 6250 tiles

  const float* xcur = x;
  for (int l = 0; l < 3; ++l) {
    zero_kernel<<<zblocks, 256, 0, stream>>>((float4*)agg, n4);
    edge_kernel<<<eblocks, 256, 0, stream>>>(xcur, ei, ea,
                                             We + (size_t)l * ED * CH,
                                             be + (size_t)l * CH, agg);
    float* xnext = (l & 1) ? bufB : bufA;
    mlp_kernel<<<nblocks, 128, 0, stream>>>(
        xcur, agg, W1 + (size_t)l * CH * CH, b1 + (size_t)l * CH,
        W2 + (size_t)l * CH * CH, b2 + (size_t)l * CH, eps + l,
        gamma + (size_t)l * CH, beta + (size_t)l * CH, xnext);
    xcur = xnext;
  }
  head_kernel<<<nblocks, 128, 0, stream>>>(xcur, Wh1, bh1, Wh2, bh2, out);
}